// MyVGG_83253646066027
// MI455X (gfx1250) — compile-verified
//
#include <hip/hip_runtime.h>

#define B_  1024
#define G_  2000
#define S_  32
#define M1_ 4096
#define M2_ 2048
#define M3_ 1024
#define H_  64
#define KCH 512   // K-chunk per GEMM workgroup

typedef float v2f __attribute__((ext_vector_type(2)));
typedef float v8f __attribute__((ext_vector_type(8)));

// ---------------------------------------------------------------------------
// Stage 1: gene[b,g] = relu(dot(x[b,g,:], Wg[g,:]) + bg[g])
// 4 lanes cooperate per gene: each lane loads 2 float4 (32B) -> the quad covers
// the gene's full 128B contiguously => perfectly coalesced global_load_b128.
// Cross-quad reduction via wave32 shfl_xor.
// ---------------------------------------------------------------------------
__global__ void gene_kernel(const float* __restrict__ x,
                            const float* __restrict__ Wg,
                            const float* __restrict__ bg,
                            float* __restrict__ gene) {
  const int b    = blockIdx.x;
  const int tid  = threadIdx.x;      // 256 threads
  const int quad = tid >> 2;         // 0..63 : gene slot
  const int sub  = tid & 3;          // 0..3  : 8-float slice within gene
  const float* __restrict__ xrow = x + (size_t)b * (G_ * S_);

  for (int g0 = 0; g0 < G_; g0 += 64) {
    const int g = g0 + quad;
    float p = 0.0f;
    if (g < G_) {
      const float4* xs = (const float4*)(xrow + (size_t)g * S_ + sub * 8);
      const float4* ws = (const float4*)(Wg   + (size_t)g * S_ + sub * 8);
      float4 xa = xs[0], xb = xs[1];
      float4 wa = ws[0], wb = ws[1];
      p = xa.x*wa.x + xa.y*wa.y + xa.z*wa.z + xa.w*wa.w
        + xb.x*wb.x + xb.y*wb.y + xb.z*wb.z + xb.w*wb.w;
    }
    // reduce across the 4-lane quad (wave32)
    p += __shfl_xor(p, 1, 32);
    p += __shfl_xor(p, 2, 32);
    if (g < G_ && sub == 0) {
      float v = p + bg[g];
      gene[(size_t)b * G_ + g] = v > 0.0f ? v : 0.0f;
    }
  }
}

// ---------------------------------------------------------------------------
// Stage 2: partial GEMMs  P[mat][chunk] += A_mat[16 rows, Kchunk] @ W_mat[., 64]
// One workgroup = 4 waves; wave w owns N-tile n0=16*w of the 64 output cols.
// grid = (64 Mtiles, 4 matrices, 8 kchunks); inactive kchunks return early
// (uniform per block -> EXEC stays all-ones around WMMA as required).
// Uses V_WMMA_F32_16X16X4_F32 (f32 A/B/C/D -> matches reference numerics).
// ---------------------------------------------------------------------------
__global__ void gemm_kernel(const float* __restrict__ gene,
                            const float* __restrict__ z1,
                            const float* __restrict__ z2,
                            const float* __restrict__ z3,
                            const float* __restrict__ W0,
                            const float* __restrict__ W1,
                            const float* __restrict__ W2,
                            const float* __restrict__ W3,
                            float* __restrict__ Pbuf) {
  const int mtile = blockIdx.x;   // 0..63  (M = 16*mtile)
  const int mat   = blockIdx.y;   // 0..3
  const int chunk = blockIdx.z;   // 0..7

  const float* A; const float* W; int Kdim;
  if      (mat == 0) { A = gene; W = W0; Kdim = G_;  }
  else if (mat == 1) { A = z1;   W = W1; Kdim = M1_; }
  else if (mat == 2) { A = z2;   W = W2; Kdim = M2_; }
  else               { A = z3;   W = W3; Kdim = M3_; }

  const int k_start = chunk * KCH;
  if (k_start >= Kdim) return;                    // uniform per workgroup
  const int k_end = (k_start + KCH < Kdim) ? (k_start + KCH) : Kdim;

  const int wave = threadIdx.x >> 5;              // 0..3 -> N-tile
  const int lane = threadIdx.x & 31;
  const int l15  = lane & 15;                     // M-index for A, N-index for B/D
  const int half = lane >> 4;                     // selects K pair
  const int m0   = mtile * 16;
  const int n0   = wave * 16;

  const float* __restrict__ Arow = A + (size_t)(m0 + l15) * Kdim;

  v8f acc = {};
  #pragma unroll 4
  for (int k = k_start; k < k_end; k += 4) {
    const int kk = k + half * 2;
    v2f a, b;
    a[0] = Arow[kk];
    a[1] = Arow[kk + 1];
    b[0] = W[(size_t)kk       * H_ + n0 + l15];
    b[1] = W[(size_t)(kk + 1) * H_ + n0 + l15];
    acc = __builtin_amdgcn_wmma_f32_16x16x4_f32(
        /*neg_a=*/false, a, /*neg_b=*/false, b,
        /*c_mod=*/(short)0, acc, /*reuse_a=*/false, /*reuse_b=*/false);
  }

  // D layout: VGPR r -> row (8*half + r), col l15 (within the 16x16 tile)
  float* __restrict__ P =
      Pbuf + ((size_t)(mat * 8 + chunk) * B_ + m0) * H_ + n0;
  #pragma unroll
  for (int r = 0; r < 8; ++r) {
    const int row = half * 8 + r;
    P[(size_t)row * H_ + l15] = acc[r];
  }
}

// ---------------------------------------------------------------------------
// Stage 3: sum K-chunk partials, per-matrix ReLU + learned-weight fusion,
// then the 64->64->64->1 MLP. One 64-thread workgroup per batch row.
// ---------------------------------------------------------------------------
__global__ void combine_kernel(const float* __restrict__ Pbuf,
                               const float* __restrict__ b0,
                               const float* __restrict__ b1,
                               const float* __restrict__ b2,
                               const float* __restrict__ b3,
                               const float* __restrict__ w1,
                               const float* __restrict__ w2,
                               const float* __restrict__ w3,
                               const float* __restrict__ fW1,
                               const float* __restrict__ fb1,
                               const float* __restrict__ fW2,
                               const float* __restrict__ fb2,
                               const float* __restrict__ fW3,
                               const float* __restrict__ fb3,
                               float* __restrict__ out) {
  __shared__ float sh[H_];
  __shared__ float sh2[H_];
  const int b = blockIdx.x;
  const int t = threadIdx.x;   // 0..63

  const int chunks[4] = {4, 8, 4, 2};   // ceil(K/512) for gene,z1,z2,z3
  float sums[4];
  #pragma unroll
  for (int m = 0; m < 4; ++m) {
    float s = 0.0f;
    for (int c = 0; c < chunks[m]; ++c)
      s += Pbuf[((size_t)(m * 8 + c) * B_ + b) * H_ + t];
    sums[m] = s;
  }

  float a0 = sums[0] + b0[t]; a0 = a0 > 0.0f ? a0 : 0.0f;
  float a1 = sums[1] + b1[t]; a1 = a1 > 0.0f ? a1 : 0.0f;
  float a2 = sums[2] + b2[t]; a2 = a2 > 0.0f ? a2 : 0.0f;
  float a3 = sums[3] + b3[t]; a3 = a3 > 0.0f ? a3 : 0.0f;
  float d3 = a0 + a1 * w1[t] + a2 * w2[t] + a3 * w3[t];

  sh[t] = d3;
  __syncthreads();
  float h1 = fb1[t];
  #pragma unroll 8
  for (int i = 0; i < H_; ++i) h1 += sh[i] * fW1[i * H_ + t];
  h1 = h1 > 0.0f ? h1 : 0.0f;
  sh2[t] = h1;
  __syncthreads();
  float h2 = fb2[t];
  #pragma unroll 8
  for (int i = 0; i < H_; ++i) h2 += sh2[i] * fW2[i * H_ + t];
  h2 = h2 > 0.0f ? h2 : 0.0f;
  __syncthreads();
  sh[t] = h2 * fW3[t];
  __syncthreads();
  if (t == 0) {
    float s = fb3[0];
    for (int i = 0; i < H_; ++i) s += sh[i];
    out[b] = s;
  }
}

// ---------------------------------------------------------------------------
extern "C" void kernel_launch(void* const* d_in, const int* in_sizes, int n_in,
                              void* d_out, int out_size, void* d_ws, size_t ws_size,
                              hipStream_t stream) {
  const float* x   = (const float*)d_in[0];
  const float* z1  = (const float*)d_in[1];
  const float* z2  = (const float*)d_in[2];
  const float* z3  = (const float*)d_in[3];
  const float* Wg  = (const float*)d_in[4];
  const float* bg  = (const float*)d_in[5];
  const float* W0  = (const float*)d_in[6];
  const float* b0  = (const float*)d_in[7];
  const float* W1  = (const float*)d_in[8];
  const float* b1  = (const float*)d_in[9];
  const float* W2  = (const float*)d_in[10];
  const float* b2  = (const float*)d_in[11];
  const float* W3  = (const float*)d_in[12];
  const float* b3  = (const float*)d_in[13];
  const float* w1  = (const float*)d_in[14];
  const float* w2  = (const float*)d_in[15];
  const float* w3  = (const float*)d_in[16];
  const float* fW1 = (const float*)d_in[17];
  const float* fb1 = (const float*)d_in[18];
  const float* fW2 = (const float*)d_in[19];
  const float* fb2 = (const float*)d_in[20];
  const float* fW3 = (const float*)d_in[21];
  const float* fb3 = (const float*)d_in[22];

  float* gene = (float*)d_ws;                                   // B*G floats = 8 MB
  float* Pbuf = (float*)((char*)d_ws + (size_t)B_ * G_ * sizeof(float)); // 32*B*H floats = 8 MB

  gene_kernel<<<B_, 256, 0, stream>>>(x, Wg, bg, gene);

  dim3 ggrid(B_ / 16, 4, 8);
  gemm_kernel<<<ggrid, 128, 0, stream>>>(gene, z1, z2, z3, W0, W1, W2, W3, Pbuf);

  combine_kernel<<<B_, H_, 0, stream>>>(Pbuf, b0, b1, b2, b3, w1, w2, w3,
                                        fW1, fb1, fW2, fb2, fW3, fb3,
                                        (float*)d_out);
}